// Model_83141976916165
// MI455X (gfx1250) — compile-verified
//
#include <hip/hip_runtime.h>
#include <cmath>

// ---------------------------------------------------------------------------
// Bidirectional 6-layer GRU for MI455X (gfx1250), fp32 via V_WMMA_F32_16X16X4_F32.
// T=512, B=64, I=128, H=256, L=6.  d_out = 12 final hidden states (12,64,256).
// ---------------------------------------------------------------------------

typedef float v2f __attribute__((ext_vector_type(2)));
typedef float v8f __attribute__((ext_vector_type(8)));

#define T_LEN 512
#define BATCH 64
#define HID   256
#define G3    768          // 3*H gate width
#define NLAY  6
#define NBLK  8            // blocks per direction in the recurrence kernel

__device__ __forceinline__ float sigmoidf_(float x) {
    return 1.0f / (1.0f + __expf(-x));
}

// ---------------------------------------------------------------------------
// Weight pre-transpose: w (N=768 rows, K cols, row-major) -> K-pair-packed
// B-matrix layout: out[p*768 + g] = float2( w[g, 2p], w[g, 2p+1] ).
// One 8-byte load then yields the (V0,V1) B-fragment pair for WMMA.
// ---------------------------------------------------------------------------
struct PrepArgs {
    const float* w_ih[12];
    const float* w_hh[12];
    float2*      wiT2[12];
    float2*      whT2[12];
    int          K[12];
};

__global__ __launch_bounds__(256)
void prep_transpose_kernel(PrepArgs P)
{
    const int ld  = blockIdx.y;
    const int idx = blockIdx.x * 256 + threadIdx.x;
    if (blockIdx.z == 0) {
        const int K = P.K[ld];
        const int n = (K / 2) * G3;
        if (idx < n) {
            const int p = idx / G3, g = idx % G3;
            const float* __restrict__ w = P.w_ih[ld];
            float2 v;
            v.x = w[(long long)g * K + 2 * p];
            v.y = w[(long long)g * K + 2 * p + 1];
            P.wiT2[ld][idx] = v;
        }
    } else {
        const int n = (HID / 2) * G3;
        if (idx < n) {
            const int p = idx / G3, g = idx % G3;
            const float* __restrict__ w = P.w_hh[ld];
            float2 v;
            v.x = w[g * HID + 2 * p];
            v.y = w[g * HID + 2 * p + 1];
            P.whT2[ld][idx] = v;
        }
    }
}

// ---------------------------------------------------------------------------
// Input projection: gi[dir] = layer_in @ w_ih[dir]^T + b_ih[dir]
//   A: (T*B=32768, K) row-major, BT: K-pair-packed (K x 768), gi: (32768, 768).
// Block = 256 thr = 8 waves; block tile 128(M) x 64(N); wave tile 16 x 64
// (4 WMMA N-tiles sharing one A-fragment per K-step of 4).
// ---------------------------------------------------------------------------
__global__ __launch_bounds__(256, 1)
void gi_gemm_kernel(const float* __restrict__ A, int K,
                    const float2* __restrict__ BT_f, const float2* __restrict__ BT_b,
                    const float* __restrict__ bias_f, const float* __restrict__ bias_b,
                    float* __restrict__ gi_base, long long giStride)
{
    const int dir = blockIdx.z;
    const float2* __restrict__ BT   = dir ? BT_b   : BT_f;
    const float*  __restrict__ bias = dir ? bias_b : bias_f;
    float* __restrict__ gi = gi_base + (long long)dir * giStride;

    const int tid  = threadIdx.x;
    const int wv   = tid >> 5;
    const int lane = tid & 31;
    const int half = lane >> 4;       // 0: lanes 0-15 (K,K+1), 1: lanes 16-31 (K+2,K+3)
    const int l16  = lane & 15;
    const int m0   = blockIdx.x * 128 + wv * 16;
    const int n0   = blockIdx.y * 64;

    v8f c0 = {}, c1 = {}, c2 = {}, c3 = {};
    const float* __restrict__ arow = A + (long long)(m0 + l16) * K;

    for (int k0 = 0; k0 < K; k0 += 4) {
        const float2 af = *(const float2*)(arow + k0 + 2 * half);
        v2f a; a.x = af.x; a.y = af.y;
        const float2* __restrict__ bp =
            BT + (long long)((k0 >> 1) + half) * G3 + n0 + l16;
        const float2 f0 = bp[0], f1 = bp[16], f2 = bp[32], f3 = bp[48];
        v2f b0; b0.x = f0.x; b0.y = f0.y;
        v2f b1; b1.x = f1.x; b1.y = f1.y;
        v2f b2; b2.x = f2.x; b2.y = f2.y;
        v2f b3; b3.x = f3.x; b3.y = f3.y;
        c0 = __builtin_amdgcn_wmma_f32_16x16x4_f32(false, a, false, b0, (short)0, c0, false, false);
        c1 = __builtin_amdgcn_wmma_f32_16x16x4_f32(false, a, false, b1, (short)0, c1, false, false);
        c2 = __builtin_amdgcn_wmma_f32_16x16x4_f32(false, a, false, b2, (short)0, c2, false, false);
        c3 = __builtin_amdgcn_wmma_f32_16x16x4_f32(false, a, false, b3, (short)0, c3, false, false);
    }

    // C layout: VGPR v -> row m0+v (lanes 0-15) / m0+v+8 (lanes 16-31), col = lane%16.
    v8f cc[4] = {c0, c1, c2, c3};
    #pragma unroll
    for (int j = 0; j < 4; ++j) {
        const int   col = n0 + 16 * j + l16;
        const float bv  = bias[col];
        #pragma unroll
        for (int v = 0; v < 8; ++v) {
            const int row = m0 + v + half * 8;
            gi[(long long)row * G3 + col] = cc[j][v] + bv;
        }
    }
}

// ---------------------------------------------------------------------------
// Persistent recurrence kernel for one layer, both directions.
// grid = 16 blocks: blocks 0-7 forward, 8-15 backward; each block owns 32 of
// the 256 h-columns (-> 96 gate columns).  Per step: 64x96x256 fp32-WMMA GEMM
// (wave w: M-tile w/2, N-tiles 3*(w&1)+{0,1,2}; 64 K-steps), gate exchange via
// LDS, h ping-pong in global ws, device-scope spin barrier between steps.
// ---------------------------------------------------------------------------
__global__ __launch_bounds__(256, 1)
void gru_recur_kernel(const float2* __restrict__ WT_f, const float2* __restrict__ WT_b,
                      const float* __restrict__ bhh_f, const float* __restrict__ bhh_b,
                      const float* __restrict__ gi_base, long long giStride,
                      const float* __restrict__ h0,
                      float* __restrict__ hbuf,      // 2 dirs x 2 (ping/pong) x 64x256
                      float* __restrict__ out,       // (T, B, 2H) concat output
                      float* __restrict__ hfinal,    // d_out (12, B, H)
                      unsigned int* __restrict__ counters,
                      int layer)
{
    const int dir = blockIdx.x >> 3;
    const int blk = blockIdx.x & 7;
    const float2* __restrict__ WT  = dir ? WT_b  : WT_f;
    const float*  __restrict__ bhh = dir ? bhh_b : bhh_f;
    const float*  __restrict__ gi  = gi_base + (long long)dir * giStride;
    unsigned int* __restrict__ cnt = counters + (layer * 2 + dir);
    float* __restrict__ hping = hbuf + dir * 2 * BATCH * HID;

    const int tid   = threadIdx.x;
    const int wv    = tid >> 5;
    const int lane  = tid & 31;
    const int half  = lane >> 4;
    const int l16   = lane & 15;
    const int b32   = blk * 32;
    const int m0    = (wv >> 1) * 16;       // M-tile of this wave
    const int jbase = (wv & 1) * 3;         // first of 3 N-tiles of this wave

    __shared__ float lds_gh[BATCH * 96];    // 24 KB gate-slice exchange buffer

    // Initialize h ping buffer from h0 slice (this block's 32 columns).
    const float* __restrict__ h0s = h0 + (long long)(2 * layer + dir) * BATCH * HID;
    #pragma unroll
    for (int i = 0; i < 8; ++i) {
        const int idx = tid + 256 * i;
        const int row = idx >> 5;
        const int col = (idx & 31) + b32;
        hping[row * HID + col] = h0s[row * HID + col];
    }

    unsigned int round = 1;
    __threadfence();
    __syncthreads();
    if (tid == 0) {
        __hip_atomic_fetch_add(cnt, 1u, __ATOMIC_ACQ_REL, __HIP_MEMORY_SCOPE_AGENT);
        while (__hip_atomic_load(cnt, __ATOMIC_ACQUIRE, __HIP_MEMORY_SCOPE_AGENT) < 8u * round)
            __builtin_amdgcn_s_sleep(2);
    }
    __syncthreads();

    // Global gate column (in w_hh row space) for each of this wave's 3 N-tiles.
    int gcol[3];
    #pragma unroll
    for (int q = 0; q < 3; ++q) {
        const int j = jbase + q;
        gcol[q] = (j >> 1) * HID + b32 + (j & 1) * 16 + l16;
    }

    for (int s = 0; s < T_LEN; ++s) {
        const int t = dir ? (T_LEN - 1 - s) : s;
        const float* __restrict__ hcur = hping + (s & 1) * BATCH * HID;
        float*       __restrict__ hnxt = hping + ((s + 1) & 1) * BATCH * HID;

        // ---- gh-slice GEMM: (16 x 48) per wave, K = 256 ----
        v8f c0 = {}, c1 = {}, c2 = {};
        const float* __restrict__ arow = hcur + (m0 + l16) * HID;
        for (int k0 = 0; k0 < HID; k0 += 4) {
            const float2 af = *(const float2*)(arow + k0 + 2 * half);
            v2f a; a.x = af.x; a.y = af.y;
            const float2* __restrict__ bp = WT + ((k0 >> 1) + half) * G3;
            const float2 f0 = bp[gcol[0]], f1 = bp[gcol[1]], f2 = bp[gcol[2]];
            v2f b0; b0.x = f0.x; b0.y = f0.y;
            v2f b1; b1.x = f1.x; b1.y = f1.y;
            v2f b2; b2.x = f2.x; b2.y = f2.y;
            c0 = __builtin_amdgcn_wmma_f32_16x16x4_f32(false, a, false, b0, (short)0, c0, false, false);
            c1 = __builtin_amdgcn_wmma_f32_16x16x4_f32(false, a, false, b1, (short)0, c1, false, false);
            c2 = __builtin_amdgcn_wmma_f32_16x16x4_f32(false, a, false, b2, (short)0, c2, false, false);
        }

        // Scatter C tiles into LDS gate slice (64 rows x 96 cols: r|z|n).
        v8f cc[3] = {c0, c1, c2};
        #pragma unroll
        for (int q = 0; q < 3; ++q) {
            const int scol = 16 * (jbase + q) + l16;
            #pragma unroll
            for (int v = 0; v < 8; ++v)
                lds_gh[(m0 + v + half * 8) * 96 + scol] = cc[q][v];
        }
        __syncthreads();

        // ---- gate math + state update for this block's 32 h-columns ----
        const float* __restrict__ git = gi + (long long)t * BATCH * G3;
        #pragma unroll
        for (int i = 0; i < 8; ++i) {
            const int idx = tid + 256 * i;
            const int row = idx >> 5;
            const int col = idx & 31;
            const int g   = b32 + col;
            const float* __restrict__ gr = git + row * G3;
            const float ghr = lds_gh[row * 96 +       col] + bhh[g];
            const float ghz = lds_gh[row * 96 + 32 +  col] + bhh[HID + g];
            const float ghn = lds_gh[row * 96 + 64 +  col] + bhh[2 * HID + g];
            const float r = sigmoidf_(gr[g] + ghr);
            const float z = sigmoidf_(gr[HID + g] + ghz);
            const float n = tanhf(gr[2 * HID + g] + r * ghn);
            const float hp = hcur[row * HID + g];
            const float hn = (1.0f - z) * n + z * hp;
            hnxt[row * HID + g] = hn;
            out[(long long)t * BATCH * 2 * HID + row * 2 * HID + dir * HID + g] = hn;
            if (s == T_LEN - 1)
                hfinal[(long long)(2 * layer + dir) * BATCH * HID + row * HID + g] = hn;
        }

        // ---- device-scope barrier (per direction) before next step ----
        ++round;
        __threadfence();
        __syncthreads();
        if (tid == 0) {
            __hip_atomic_fetch_add(cnt, 1u, __ATOMIC_ACQ_REL, __HIP_MEMORY_SCOPE_AGENT);
            while (__hip_atomic_load(cnt, __ATOMIC_ACQUIRE, __HIP_MEMORY_SCOPE_AGENT) < 8u * round)
                __builtin_amdgcn_s_sleep(2);
        }
        __syncthreads();
    }
}

// ---------------------------------------------------------------------------
// Host orchestration (graph-capture safe: only kernel launches + memsetAsync).
// ---------------------------------------------------------------------------
extern "C" void kernel_launch(void* const* d_in, const int* in_sizes, int n_in,
                              void* d_out, int out_size, void* d_ws, size_t ws_size,
                              hipStream_t stream)
{
    (void)in_sizes; (void)n_in; (void)out_size; (void)ws_size;

    const float* x  = (const float*)d_in[0];
    const float* h0 = (const float*)d_in[1];
    const float* w_ih[12]; const float* w_hh[12];
    const float* b_ih[12]; const float* b_hh[12];
    for (int ld = 0; ld < 12; ++ld) {
        w_ih[ld] = (const float*)d_in[2 + ld * 4 + 0];
        w_hh[ld] = (const float*)d_in[2 + ld * 4 + 1];
        b_ih[ld] = (const float*)d_in[2 + ld * 4 + 2];
        b_hh[ld] = (const float*)d_in[2 + ld * 4 + 3];
    }

    char*  ws  = (char*)d_ws;
    size_t off = 0;
    auto alloc = [&](size_t bytes) -> void* {
        void* p = ws + off;
        off = (off + bytes + 255) & ~(size_t)255;
        return p;
    };

    unsigned int* counters = (unsigned int*)alloc(64 * sizeof(unsigned int));

    PrepArgs pa;
    float2* wiT2[12]; float2* whT2[12];
    for (int ld = 0; ld < 12; ++ld) {
        const int K = (ld < 2) ? 128 : 512;   // layer 0: I=128, layers 1-5: 2H=512
        wiT2[ld] = (float2*)alloc((size_t)(K / 2) * G3 * sizeof(float2));
        whT2[ld] = (float2*)alloc((size_t)(HID / 2) * G3 * sizeof(float2));
        pa.w_ih[ld] = w_ih[ld]; pa.w_hh[ld] = w_hh[ld];
        pa.wiT2[ld] = wiT2[ld]; pa.whT2[ld] = whT2[ld];
        pa.K[ld] = K;
    }

    const long long giStride = (long long)T_LEN * BATCH * G3;       // per direction
    float* gi   = (float*)alloc(2 * (size_t)giStride * sizeof(float));
    float* bufA = (float*)alloc((size_t)T_LEN * BATCH * 2 * HID * sizeof(float));
    float* bufB = (float*)alloc((size_t)T_LEN * BATCH * 2 * HID * sizeof(float));
    float* hbuf = (float*)alloc(4 * (size_t)BATCH * HID * sizeof(float));

    hipMemsetAsync(counters, 0, 64 * sizeof(unsigned int), stream);

    // Pre-transpose all weights into WMMA-friendly K-pair-packed layout.
    prep_transpose_kernel<<<dim3(768, 12, 2), 256, 0, stream>>>(pa);

    const float* lin = x;
    float* bufs[2] = {bufA, bufB};
    for (int l = 0; l < NLAY; ++l) {
        const int K   = (l == 0) ? 128 : 512;
        const int ldf = 2 * l, ldb = 2 * l + 1;

        // gi = lin @ w_ih^T + b_ih   (both directions in one grid, z = dir)
        dim3 g1((T_LEN * BATCH) / 128, G3 / 64, 2);
        gi_gemm_kernel<<<g1, 256, 0, stream>>>(lin, K,
                                               wiT2[ldf], wiT2[ldb],
                                               b_ih[ldf], b_ih[ldb],
                                               gi, giStride);

        // Persistent bidirectional scan for this layer.
        float* lout = bufs[l & 1];
        gru_recur_kernel<<<2 * NBLK, 256, 0, stream>>>(whT2[ldf], whT2[ldb],
                                                       b_hh[ldf], b_hh[ldb],
                                                       gi, giStride,
                                                       h0, hbuf, lout,
                                                       (float*)d_out,
                                                       counters, l);
        lin = lout;
    }
}